// SparseAttention_44315472560500
// MI455X (gfx1250) — compile-verified
//
#include <hip/hip_runtime.h>

typedef __attribute__((ext_vector_type(2))) float v2f;
typedef __attribute__((ext_vector_type(4))) float v4f;
typedef __attribute__((ext_vector_type(8))) float v8f;

#define HH 8
#define SS 4096
#define DD 64
#define BS 64
#define NB (SS / BS)          // 64 query blocks
#define W3 (3 * BS)           // 192-wide local window
#define KSTRIDE 68            // LDS row stride for K window (bank stride 4 -> conflict free)
#define WSTRIDE 196           // LDS row stride for W tiles   (bank stride 4 -> conflict free)
#define SMEM_FLOATS (W3 * KSTRIDE)            // 13,056 floats = 52,224 B (also >= 4*16*WSTRIDE)
#define OUT_ELEMS (HH * SS * DD)              // 2,097,152 floats
#define DENSE_ELEMS ((long)HH * SS * SS)      // 134,217,728 floats

// ---------------------------------------------------------------------------
// Kernel 1: zero the 537 MB dense attention-weights region. Non-temporal
// native-vector stores: dense >> 192 MB L2, keep Q/K/V (24 MB) resident.
// ---------------------------------------------------------------------------
__global__ void zero_dense_kernel(v4f* __restrict__ p, long n4) {
    long i = (long)blockIdx.x * blockDim.x + threadIdx.x;
    long stride = (long)gridDim.x * blockDim.x;
    v4f z = (v4f){0.f, 0.f, 0.f, 0.f};
    for (; i < n4; i += stride) __builtin_nontemporal_store(z, &p[i]);
}

// ---------------------------------------------------------------------------
// Kernel 2: block-local attention. One workgroup per (query-block, head).
// 4 wave32s; wave w owns query rows [qb*64 + 16w, +16).
//   stage K window -> LDS; Phase 1: S = (Q Kt)/8 via v_wmma_f32_16x16x4_f32
//   (A from global, B from LDS, batched 12-frag loads); softmax;
//   reuse LDS for W; Phase 2: O = W V (A from LDS, B from global, batched).
// ---------------------------------------------------------------------------
__global__ __launch_bounds__(128) void blockattn_kernel(
    const float* __restrict__ q, const float* __restrict__ k,
    const float* __restrict__ v, float* __restrict__ out,
    float* __restrict__ dense)
{
    __shared__ float smem[SMEM_FLOATS];   // K window, later recycled for W tiles

    const int qb   = blockIdx.x;
    const int h    = blockIdx.y;
    const int tid  = threadIdx.x;
    const int lane = tid & 31;
    const int wave = tid >> 5;
    const int half = lane >> 4;      // 0: lanes 0-15, 1: lanes 16-31
    const int l16  = lane & 15;
    const int koff = 2 * half;       // WMMA f32 16x16x4: lanes>=16 hold K={2,3}

    const int win0  = qb * BS - BS;          // global column of window pos 0
    const int qrow0 = qb * BS + wave * 16;   // this wave's first query row

    const float* qh = q + (size_t)h * SS * DD;
    const float* kh = k + (size_t)h * SS * DD;
    const float* vh = v + (size_t)h * SS * DD;

    // Prefetch the V window (used in phase 2) while phase 1 runs.
    {
        int r0 = win0 + tid;
        if (r0 >= 0 && r0 < SS) __builtin_prefetch(vh + (size_t)r0 * DD, 0, 0);
        int r1 = win0 + tid + 128;
        if (tid < 64 && r1 >= 0 && r1 < SS) __builtin_prefetch(vh + (size_t)r1 * DD, 0, 0);
    }

    // ---- cooperative stage of K window: smem[r*68 + c], r in [0,192) ----
    {
#pragma unroll
        for (int i = 0; i < 24; ++i) {
            int idx = tid + i * 128;          // 0..3071 float4 tiles
            int r   = idx >> 4;               // window row 0..191
            int c   = (idx & 15) << 2;        // col 0,4,..,60
            int gr  = win0 + r;
            gr = gr < 0 ? 0 : (gr > SS - 1 ? SS - 1 : gr);  // clamped rows masked later
            v4f val = *(const v4f*)(kh + (size_t)gr * DD + c);
            *(v4f*)(&smem[r * KSTRIDE + c]) = val;
        }
    }
    __syncthreads();

    // ---------------- Phase 1: scores = Q * K^T ----------------
    v8f acc[12];
#pragma unroll
    for (int t = 0; t < 12; ++t) acc[t] = (v8f){0.f,0.f,0.f,0.f,0.f,0.f,0.f,0.f};

    const float* qp = qh + (size_t)(qrow0 + l16) * DD + koff;  // imm offsets: kk*4
    const float* kb = smem + l16 * KSTRIDE + koff;             // imm offsets: t*4352+kk*4

#pragma unroll
    for (int kk = 0; kk < DD; kk += 4) {
        v2f a = *(const v2f*)(qp + kk);
        v2f bf[12];
#pragma unroll
        for (int t = 0; t < 12; ++t)
            bf[t] = *(const v2f*)(kb + t * 16 * KSTRIDE + kk);
#pragma unroll
        for (int t = 0; t < 12; ++t)
            acc[t] = __builtin_amdgcn_wmma_f32_16x16x4_f32(
                false, a, false, bf[t], (short)0, acc[t], false, false);
    }

    // scale + mask invalid window columns
    const float NEG_INF = -__builtin_inff();
#pragma unroll
    for (int t = 0; t < 12; ++t) {
        int gc = win0 + t * 16 + l16;
        bool valid = (gc >= 0) && (gc < SS);
#pragma unroll
        for (int r = 0; r < 8; ++r)
            acc[t][r] = valid ? acc[t][r] * 0.125f : NEG_INF;
    }

    // softmax over the 192 columns of each row. Row m of a C tile lives in
    // VGPR m%8 across the 16 lanes of half m/8 -> shfl_xor within halves.
#pragma unroll
    for (int r = 0; r < 8; ++r) {
        float m = acc[0][r];
#pragma unroll
        for (int t = 1; t < 12; ++t) m = fmaxf(m, acc[t][r]);
        m = fmaxf(m, __shfl_xor(m, 1, 32));
        m = fmaxf(m, __shfl_xor(m, 2, 32));
        m = fmaxf(m, __shfl_xor(m, 4, 32));
        m = fmaxf(m, __shfl_xor(m, 8, 32));
        float s = 0.f;
#pragma unroll
        for (int t = 0; t < 12; ++t) {
            float p = __expf(acc[t][r] - m);        // exp(-inf)=0 at masked cols
            acc[t][r] = p;
            s += p;
        }
        s += __shfl_xor(s, 1, 32);
        s += __shfl_xor(s, 2, 32);
        s += __shfl_xor(s, 4, 32);
        s += __shfl_xor(s, 8, 32);
        float is = __builtin_amdgcn_rcpf(s);
#pragma unroll
        for (int t = 0; t < 12; ++t) acc[t][r] *= is;
    }

    // All waves finished reading K from LDS -> recycle smem for W tiles.
    __syncthreads();

    float* wl = smem + wave * 16 * WSTRIDE;        // this wave's 16x196 W tile
    float* dh = dense + (size_t)h * SS * SS;
#pragma unroll
    for (int r = 0; r < 8; ++r) {
        int row  = r + 8 * half;                   // row within 16-row tile
        int grow = qrow0 + row;
#pragma unroll
        for (int t = 0; t < 12; ++t) {
            float wv = acc[t][r];
            wl[row * WSTRIDE + t * 16 + l16] = wv;
            int gc = win0 + t * 16 + l16;
            if (gc >= 0 && gc < SS)
                __builtin_nontemporal_store(wv, &dh[(size_t)grow * SS + gc]);
        }
    }

    // ---------------- Phase 2: out = W * V ----------------
    v8f oacc[4];
#pragma unroll
    for (int t = 0; t < 4; ++t) oacc[t] = (v8f){0.f,0.f,0.f,0.f,0.f,0.f,0.f,0.f};

    const float* wb = wl + l16 * WSTRIDE + koff;   // imm offsets: kk*4

#pragma unroll 8
    for (int kk = 0; kk < W3; kk += 4) {
        v2f a = *(const v2f*)(wb + kk);            // W[m][kk+koff .. +1] from LDS
        int vr0 = win0 + kk + koff;
        int vr1 = vr0 + 1;
        int c0 = vr0 < 0 ? 0 : (vr0 > SS - 1 ? SS - 1 : vr0);  // clamp: W=0 there
        int c1 = vr1 < 0 ? 0 : (vr1 > SS - 1 ? SS - 1 : vr1);
        const float* vp0 = vh + (size_t)c0 * DD + l16;
        const float* vp1 = vh + (size_t)c1 * DD + l16;
        float bx[4], by[4];
#pragma unroll
        for (int t = 0; t < 4; ++t) { bx[t] = vp0[t * 16]; by[t] = vp1[t * 16]; }
#pragma unroll
        for (int t = 0; t < 4; ++t) {
            v2f b; b.x = bx[t]; b.y = by[t];
            oacc[t] = __builtin_amdgcn_wmma_f32_16x16x4_f32(
                false, a, false, b, (short)0, oacc[t], false, false);
        }
    }

    // store O tile
    float* oh = out + (size_t)h * SS * DD;
#pragma unroll
    for (int r = 0; r < 8; ++r) {
        int grow = qrow0 + r + 8 * half;
#pragma unroll
        for (int t = 0; t < 4; ++t)
            oh[(size_t)grow * DD + t * 16 + l16] = oacc[t][r];
    }
}

extern "C" void kernel_launch(void* const* d_in, const int* in_sizes, int n_in,
                              void* d_out, int out_size, void* d_ws, size_t ws_size,
                              hipStream_t stream) {
    const float* q = (const float*)d_in[0];
    const float* k = (const float*)d_in[1];
    const float* v = (const float*)d_in[2];
    float* out   = (float*)d_out;
    float* dense = out + OUT_ELEMS;   // tuple layout: [out | dense], 16B-aligned offset

    // 1) zero the dense weights region (band gets overwritten below)
    zero_dense_kernel<<<4096, 256, 0, stream>>>((v4f*)dense, DENSE_ELEMS / 4);

    // 2) block-local attention: grid = (64 query blocks, 8 heads), 4 wave32s each
    dim3 grid(NB, HH);
    blockattn_kernel<<<grid, 128, 0, stream>>>(q, k, v, out, dense);
}